// ConvLayer_55551107007158
// MI455X (gfx1250) — compile-verified
//
#include <hip/hip_runtime.h>

typedef __attribute__((ext_vector_type(2))) float v2f;
typedef __attribute__((ext_vector_type(8))) float v8f;

#define DIN 128
#define NTILE 8        // 128 / 16 output column tiles
#define KPANEL 32      // K panel staged in LDS
#define KSTRIDE 34     // padded LDS row stride (bank-conflict free, 8B aligned)

__global__ __launch_bounds__(256) void zero_ws_kernel(float* ws, int n) {
    int i = blockIdx.x * 256 + threadIdx.x;
    if (i < n) ws[i] = 0.0f;
}

// One wave32 per edge: 32 lanes x float4 = full 512B row, coalesced gather,
// then 4 global_atomic_add_f32 per lane into the dst accumulator (L2-resident).
__global__ __launch_bounds__(256) void edge_scatter_kernel(
    const float* __restrict__ h_neigh, const int* __restrict__ src,
    const int* __restrict__ dst, float* aggsum, float* deg, int E) {
    int edge = blockIdx.x * 8 + (threadIdx.x >> 5);
    if (edge >= E) return;
    int lane = threadIdx.x & 31;
    int s = src[edge];
    int d = dst[edge];
    const float4* row = (const float4*)(h_neigh + (size_t)s * DIN);
    float4 v = row[lane];
    float* o = aggsum + (size_t)d * DIN + lane * 4;
    atomicAdd(o + 0, v.x);
    atomicAdd(o + 1, v.y);
    atomicAdd(o + 2, v.z);
    atomicAdd(o + 3, v.w);
    if (lane == 0) atomicAdd(deg + d, 1.0f);
}

// z = relu(h_self*Wself^T + mean_agg*Wneigh^T), then L2-normalize rows.
// One wave per 16-row M tile; V_WMMA_F32_16X16X4_F32 over K=128 (32 steps).
__global__ __launch_bounds__(128) void gemm_relu_norm_kernel(
    const float* __restrict__ h_self, const float* __restrict__ aggsum,
    const float* __restrict__ deg, const float* __restrict__ Wself,
    const float* __restrict__ Wneigh, float* __restrict__ out, int M) {
    __shared__ float sW[2][DIN][KSTRIDE];   // 2*128*34*4 = 34816 B

    const int t    = threadIdx.x;
    const int wave = t >> 5;
    const int lane = t & 31;
    const int mtiles = (M + 15) >> 4;
    const int tile = blockIdx.x * 4 + wave;
    const bool valid = tile < mtiles;
    const int m0 = tile << 4;

    const int ml = lane & 15;           // A: M row; B/C/D: N column
    const int kh = (lane >> 4) << 1;    // K offset 0 or 2 per half-wave

    v8f acc[NTILE] = {};

    float invd = 1.0f;
    const float* aSelfBase = h_self;
    const float* aAggBase  = aggsum;
    if (valid) {
        invd = 1.0f / fmaxf(deg[m0 + ml], 1.0f);
        aSelfBase = h_self + (size_t)(m0 + ml) * DIN;
        aAggBase  = aggsum + (size_t)(m0 + ml) * DIN;
    }

    for (int kb = 0; kb < DIN; kb += KPANEL) {
        // Cooperative stage of both weight K-panels into LDS.
        // 2 matrices * 128 rows * 8 float4 = 2048 float4; 128 threads * 16 iters.
        #pragma unroll
        for (int i = 0; i < 16; ++i) {
            int idx = i * 128 + t;          // 0..2047
            int w   = idx >> 10;            // matrix select
            int rem = idx & 1023;
            int row = rem >> 3;
            int c4  = (rem & 7) << 2;
            const float* W = (w == 0) ? Wself : Wneigh;
            float4 v = *(const float4*)(W + (size_t)row * DIN + kb + c4);
            float* dp = &sW[w][row][c4];
            dp[0] = v.x; dp[1] = v.y; dp[2] = v.z; dp[3] = v.w;
        }
        __syncthreads();

        if (valid) {
            #pragma unroll
            for (int ks = 0; ks < KPANEL; ks += 4) {
                int k = kb + ks + kh;       // even -> 8B aligned float2 loads
                float2 as2 = *(const float2*)(aSelfBase + k);
                float2 ag2 = *(const float2*)(aAggBase + k);
                v2f aS; aS.x = as2.x;        aS.y = as2.y;
                v2f aA; aA.x = ag2.x * invd; aA.y = ag2.y * invd;
                #pragma unroll
                for (int n = 0; n < NTILE; ++n) {
                    int col = n * 16 + ml;  // W row == output column
                    const float* bs = &sW[0][col][ks + kh];
                    const float* bn = &sW[1][col][ks + kh];
                    v2f bS; bS.x = bs[0]; bS.y = bs[1];
                    v2f bN; bN.x = bn[0]; bN.y = bn[1];
                    acc[n] = __builtin_amdgcn_wmma_f32_16x16x4_f32(
                        false, aS, false, bS, (short)0, acc[n], false, false);
                    acc[n] = __builtin_amdgcn_wmma_f32_16x16x4_f32(
                        false, aA, false, bN, (short)0, acc[n], false, false);
                }
            }
        }
        __syncthreads();
    }

    if (!valid) return;

    // ReLU
    #pragma unroll
    for (int n = 0; n < NTILE; ++n) {
        #pragma unroll
        for (int r = 0; r < 8; ++r) acc[n][r] = fmaxf(acc[n][r], 0.0f);
    }

    // Row L2 norm: C/D VGPR r holds row (m0+r) in lanes 0-15, (m0+r+8) in 16-31.
    // Butterfly over masks 1,2,4,8 reduces within each 16-lane half independently.
    #pragma unroll
    for (int r = 0; r < 8; ++r) {
        float ss = 0.0f;
        #pragma unroll
        for (int n = 0; n < NTILE; ++n) { float v = acc[n][r]; ss += v * v; }
        ss += __shfl_xor(ss, 1, 32);
        ss += __shfl_xor(ss, 2, 32);
        ss += __shfl_xor(ss, 4, 32);
        ss += __shfl_xor(ss, 8, 32);
        float scale = (ss > 0.0f) ? rsqrtf(ss) : 1.0f;
        int mrow = m0 + r + ((lane >> 4) << 3);
        float* orow = out + (size_t)mrow * DIN;
        #pragma unroll
        for (int n = 0; n < NTILE; ++n) orow[n * 16 + ml] = acc[n][r] * scale;
    }
}

extern "C" void kernel_launch(void* const* d_in, const int* in_sizes, int n_in,
                              void* d_out, int out_size, void* d_ws, size_t ws_size,
                              hipStream_t stream) {
    const float* h_neigh = (const float*)d_in[0];
    const float* h_self  = (const float*)d_in[1];
    const int*   src     = (const int*)d_in[2];
    const int*   dst     = (const int*)d_in[3];
    const float* Wself   = (const float*)d_in[4];
    const float* Wneigh  = (const float*)d_in[5];
    float* out = (float*)d_out;

    const int M = in_sizes[1] / DIN;   // n_dst = 50000
    const int E = in_sizes[2];         // 1.6M edges

    float* aggsum = (float*)d_ws;                  // [M,128]
    float* deg    = aggsum + (size_t)M * DIN;      // [M]

    int zn = M * DIN + M;
    zero_ws_kernel<<<(zn + 255) / 256, 256, 0, stream>>>(aggsum, zn);

    edge_scatter_kernel<<<(E + 7) / 8, 256, 0, stream>>>(
        h_neigh, src, dst, aggsum, deg, E);

    int mtiles = (M + 15) / 16;
    gemm_relu_norm_kernel<<<(mtiles + 3) / 4, 128, 0, stream>>>(
        h_self, aggsum, deg, Wself, Wneigh, out, M);
}